// MultiSourceAttention_88192858456749
// MI455X (gfx1250) — compile-verified
//
#include <hip/hip_runtime.h>
#include <hip/hip_bf16.h>

// MI455X / gfx1250, wave32. Whole pipeline in fp32 using V_WMMA_F32_16X16X4_F32.
// Attention stages V into LDS via the CDNA5 async global->LDS path (ASYNCcnt),
// overlapped with the Q*K^T WMMA phase.

typedef __attribute__((ext_vector_type(2))) float v2f;
typedef __attribute__((ext_vector_type(8))) float v8f;
typedef __attribute__((ext_vector_type(4))) int   v4i;

#define Bn   4
#define Tn   256
#define Mn   8
#define DIMn 256
#define Hn   8
#define DHn  64
#define INNERn 512   // Hn*DHn
#define NTOK 8192    // Bn*Tn*Mn
#define BT   1024    // Bn*Tn

__device__ __forceinline__ v8f wmma4(v2f a, v2f b, v8f c) {
    return __builtin_amdgcn_wmma_f32_16x16x4_f32(
        /*neg_a=*/false, a, /*neg_b=*/false, b,
        /*c_mod=*/(short)0, c, /*reuse_a=*/false, /*reuse_b=*/false);
}

#if __has_builtin(__builtin_amdgcn_global_load_async_to_lds_b128)
#define MSA_HAVE_ASYNC_LDS 1
#else
#define MSA_HAVE_ASYNC_LDS 0
#endif

__device__ __forceinline__ void msa_wait_async0() {
#if __has_builtin(__builtin_amdgcn_s_wait_asynccnt)
    __builtin_amdgcn_s_wait_asynccnt(0);
#else
    asm volatile("s_wait_asynccnt 0x0" ::: "memory");
#endif
}

// ---------------------------------------------------------------------------
// Kernel 1: LayerNorm over DIM=256 per token. One wave per token.
// ---------------------------------------------------------------------------
__global__ void msa_ln_kernel(const float* __restrict__ x,
                              const float* __restrict__ gamma,
                              const float* __restrict__ beta,
                              float* __restrict__ xn) {
    const int wid   = threadIdx.x >> 5;
    const int lane  = threadIdx.x & 31;
    const int token = blockIdx.x * (blockDim.x >> 5) + wid;   // 0..8191

    const float* xp = x + (size_t)token * DIMn;
    float v[8];
    float s = 0.f, ss = 0.f;
#pragma unroll
    for (int i = 0; i < 8; ++i) {
        v[i] = xp[lane * 8 + i];
        s  += v[i];
        ss += v[i] * v[i];
    }
#pragma unroll
    for (int off = 16; off > 0; off >>= 1) {
        s  += __shfl_xor(s,  off, 32);
        ss += __shfl_xor(ss, off, 32);
    }
    const float mean = s * (1.f / 256.f);
    const float var  = ss * (1.f / 256.f) - mean * mean;
    const float inv  = rsqrtf(var + 1e-5f);

    float* op = xn + (size_t)token * DIMn;
#pragma unroll
    for (int i = 0; i < 8; ++i) {
        const int d = lane * 8 + i;
        op[d] = (v[i] - mean) * inv * gamma[d] + beta[d];
    }
}

// ---------------------------------------------------------------------------
// Kernel 2: QKV projection per m: [1024 x 256] x [256 x 1536].
// One wave per 16x16 tile. Scatter into head-major Q/K/V: [bhm][t][d].
// Q is pre-scaled by DIM_HEAD^-0.5 = 0.125.
// ---------------------------------------------------------------------------
__global__ void msa_qkv_kernel(const float* __restrict__ xn,
                               const float* __restrict__ wqkv,
                               float* __restrict__ qb,
                               float* __restrict__ kb,
                               float* __restrict__ vb) {
    const int lane = threadIdx.x & 31;
    const int half = lane >> 4;
    const int l16  = lane & 15;

    const int nt = blockIdx.x;   // 0..95  (col tile of 1536)
    const int rt = blockIdx.y;   // 0..63  (row tile of 1024)
    const int m  = blockIdx.z;   // 0..7

    const int row  = rt * 16 + l16;   // bt row this lane owns for A
    const int col0 = nt * 16;

    const float* Arow = xn + ((size_t)row * Mn + m) * DIMn;
    const float* Bw   = wqkv + (size_t)m * DIMn * (3 * INNERn);

    v8f acc = {};
#pragma unroll 4
    for (int k = 0; k < DIMn; k += 4) {
        v2f a;
        const float2 av = *(const float2*)(Arow + k + 2 * half);
        a.x = av.x; a.y = av.y;
        v2f bf;
        const float* bp = Bw + (size_t)(k + 2 * half) * (3 * INNERn) + col0 + l16;
        bf.x = bp[0];
        bf.y = bp[3 * INNERn];
        acc = wmma4(a, bf, acc);
    }

    // which of q/k/v this column tile belongs to (tiles never straddle heads)
    const int sec    = col0 >> 9;        // /512
    const int within = col0 & 511;
    const int h      = within >> 6;      // head
    const int d0     = within & 63;      // d base of tile
    float* dst = (sec == 0) ? qb : ((sec == 1) ? kb : vb);
    const float scl = (sec == 0) ? 0.125f : 1.0f;

#pragma unroll
    for (int r = 0; r < 8; ++r) {
        const int orow = rt * 16 + r + 8 * half;   // bt
        const int ob   = orow >> 8;                // b
        const int ot   = orow & 255;               // t
        const int bhm  = (ob * Hn + h) * Mn + m;
        dst[((size_t)bhm * Tn + ot) * DHn + d0 + l16] = acc[r] * scl;
    }
}

// ---------------------------------------------------------------------------
// Kernel 3: attention per (b,h,m), 32 query rows per workgroup (128 thr).
// V tile (256x64 f32 = 64KB) is staged into LDS with async global->LDS
// copies issued before the Q*K^T phase (overlap); scores chunk lives in LDS;
// masked softmax in LDS; P@V reads A-fragments and V-fragments from LDS.
// Output written directly in [b][t][m][h*64+d] layout.
// ---------------------------------------------------------------------------
__global__ void msa_attn_kernel(const float* __restrict__ qb,
                                const float* __restrict__ kb,
                                const float* __restrict__ vb,
                                const unsigned char* __restrict__ mask,
                                float* __restrict__ ob) {
    __shared__ float S[32][260];       // 32 query rows x 256 keys (+pad)
    __shared__ float red[32][4];
    __shared__ float Vs[Tn * DHn];     // 64KB staged V tile

    const int tid  = threadIdx.x;    // 0..127 (4 waves)
    const int wv   = tid >> 5;
    const int lane = tid & 31;
    const int half = lane >> 4;
    const int l16  = lane & 15;

    const int chunk = blockIdx.x;        // 0..7 (32 rows each)
    const int bhm   = blockIdx.y;        // 0..255 == (b*H + h)*M + m
    const int b     = bhm >> 6;
    const int h     = (bhm >> 3) & 7;
    const int m     = bhm & 7;

    const float* Q = qb + (size_t)bhm * Tn * DHn;
    const float* K = kb + (size_t)bhm * Tn * DHn;
    const float* V = vb + (size_t)bhm * Tn * DHn;

    // ---- Stage V into LDS (async if available), overlapped with phase 1.
#if MSA_HAVE_ASYNC_LDS
#pragma unroll 4
    for (int i = tid; i < (Tn * DHn) / 4; i += 128) {   // 4096 float4s / 128 thr
        __builtin_amdgcn_global_load_async_to_lds_b128(
            (__attribute__((address_space(1))) v4i*)(V + i * 4),
            (__attribute__((address_space(3))) v4i*)(&Vs[i * 4]),
            0, 0);
    }
#else
#pragma unroll 4
    for (int i = tid; i < (Tn * DHn) / 4; i += 128) {
        const float4 t = *(const float4*)(V + i * 4);
        *(float4*)(&Vs[i * 4]) = t;
    }
#endif

    // ---- Phase 1: S = Q K^T for 32 rows. Waves {0,1}->row-tile 0, {2,3}->1;
    //      each wave covers 8 of 16 j-tiles.
    const int it = wv >> 1;
    const int i0 = chunk * 32 + it * 16;

    v2f qf[16];
#pragma unroll
    for (int kk = 0; kk < 16; ++kk) {
        const float2 t2 = *(const float2*)(Q + (size_t)(i0 + l16) * DHn + kk * 4 + 2 * half);
        qf[kk].x = t2.x; qf[kk].y = t2.y;
    }

    const int jt0 = (wv & 1) * 8;
    for (int jt = jt0; jt < jt0 + 8; ++jt) {
        v8f acc = {};
#pragma unroll
        for (int kk = 0; kk < 16; ++kk) {
            v2f bf;
            const float2 t2 = *(const float2*)(K + (size_t)(jt * 16 + l16) * DHn + kk * 4 + 2 * half);
            bf.x = t2.x; bf.y = t2.y;
            acc = wmma4(qf[kk], bf, acc);
        }
#pragma unroll
        for (int r = 0; r < 8; ++r)
            S[it * 16 + r + 8 * half][jt * 16 + l16] = acc[r];
    }
    __syncthreads();

    // ---- Phase 2: masked softmax over j (rows 0..31, 4 segments of 64).
    const int rrow = tid & 31;
    const int seg  = tid >> 5;
    const int j0   = seg * 64;

    float mx = -1e30f;
    for (int j = j0; j < j0 + 64; ++j) {
        float vS = S[rrow][j];
        if (mask[((size_t)b * Tn + j) * Mn + m]) { vS = -1e30f; S[rrow][j] = vS; }
        mx = fmaxf(mx, vS);
    }
    red[rrow][seg] = mx;
    __syncthreads();
    const float rmax = fmaxf(fmaxf(red[rrow][0], red[rrow][1]),
                             fmaxf(red[rrow][2], red[rrow][3]));
    __syncthreads();

    float sum = 0.f;
    for (int j = j0; j < j0 + 64; ++j) {
        const float e = __expf(S[rrow][j] - rmax);
        S[rrow][j] = e;
        sum += e;
    }
    red[rrow][seg] = sum;
    __syncthreads();
    const float tot = red[rrow][0] + red[rrow][1] + red[rrow][2] + red[rrow][3];
    const float inv = 1.f / tot;
    for (int j = j0; j < j0 + 64; ++j)
        S[rrow][j] *= inv;

    // V staging must be complete (ASYNCcnt drained) before any wave reads Vs.
#if MSA_HAVE_ASYNC_LDS
    msa_wait_async0();
#endif
    __syncthreads();

    // ---- Phase 3: O = P V. 2 row-tiles x 4 d-tiles = 8 tasks over 4 waves.
    const int it2 = wv >> 1;
#pragma unroll
    for (int task = 0; task < 2; ++task) {
        const int nt = (wv & 1) + task * 2;    // d tile 0..3
        v8f acc = {};
#pragma unroll 8
        for (int kk = 0; kk < 64; ++kk) {
            v2f a;
            const float2 p2 = *(const float2*)(&S[it2 * 16 + l16][kk * 4 + 2 * half]);
            a.x = p2.x; a.y = p2.y;
            v2f bf;
            const float* vp = &Vs[(kk * 4 + 2 * half) * DHn + nt * 16 + l16];
            bf.x = vp[0];
            bf.y = vp[DHn];
            acc = wmma4(a, bf, acc);
        }
#pragma unroll
        for (int r = 0; r < 8; ++r) {
            const int t = chunk * 32 + it2 * 16 + r + 8 * half;
            ob[(((size_t)b * Tn + t) * Mn + m) * INNERn + h * DHn + nt * 16 + l16] = acc[r];
        }
    }
}

// ---------------------------------------------------------------------------
// Kernel 4: output projection per m: [1024 x 512] x [512 x 256] -> d_out.
// ---------------------------------------------------------------------------
__global__ void msa_outproj_kernel(const float* __restrict__ ao,
                                   const float* __restrict__ wout,
                                   float* __restrict__ out) {
    const int lane = threadIdx.x & 31;
    const int half = lane >> 4;
    const int l16  = lane & 15;

    const int nt = blockIdx.x;   // 0..15
    const int rt = blockIdx.y;   // 0..63
    const int m  = blockIdx.z;   // 0..7

    const int row = rt * 16 + l16;
    const float* Arow = ao + ((size_t)row * Mn + m) * INNERn;
    const float* Bw   = wout + (size_t)m * INNERn * DIMn;

    v8f acc = {};
#pragma unroll 4
    for (int k = 0; k < INNERn; k += 4) {
        v2f a;
        const float2 av = *(const float2*)(Arow + k + 2 * half);
        a.x = av.x; a.y = av.y;
        v2f bf;
        const float* bp = Bw + (size_t)(k + 2 * half) * DIMn + nt * 16 + l16;
        bf.x = bp[0];
        bf.y = bp[DIMn];
        acc = wmma4(a, bf, acc);
    }

#pragma unroll
    for (int r = 0; r < 8; ++r) {
        const int orow = rt * 16 + r + 8 * half;   // bt
        out[((size_t)orow * Mn + m) * DIMn + nt * 16 + l16] = acc[r];
    }
}

// ---------------------------------------------------------------------------
extern "C" void kernel_launch(void* const* d_in, const int* in_sizes, int n_in,
                              void* d_out, int out_size, void* d_ws, size_t ws_size,
                              hipStream_t stream) {
    const float* x     = (const float*)d_in[0];
    const float* gamma = (const float*)d_in[1];
    const float* beta  = (const float*)d_in[2];
    const float* wqkv  = (const float*)d_in[3];
    const float* wout  = (const float*)d_in[4];
    const unsigned char* mask = (const unsigned char*)d_in[5];

    float* ws = (float*)d_ws;
    float* xn = ws;                               // 8192*256
    float* qb = xn + (size_t)NTOK * DIMn;         // 8192*512
    float* kb = qb + (size_t)NTOK * INNERn;
    float* vb = kb + (size_t)NTOK * INNERn;
    float* ao = vb + (size_t)NTOK * INNERn;       // 8192*512

    msa_ln_kernel<<<NTOK / 8, 256, 0, stream>>>(x, gamma, beta, xn);
    msa_qkv_kernel<<<dim3(96, 64, 8), 32, 0, stream>>>(xn, wqkv, qb, kb, vb);
    msa_attn_kernel<<<dim3(8, 256), 128, 0, stream>>>(qb, kb, vb, mask, ao);
    msa_outproj_kernel<<<dim3(16, 64, 8), 32, 0, stream>>>(ao, wout, (float*)d_out);
}